// Model_76562087018932
// MI455X (gfx1250) — compile-verified
//
#include <hip/hip_runtime.h>
#include <math.h>

// ---------------------------------------------------------------------------
// Problem constants (match reference)
// ---------------------------------------------------------------------------
#define BB_   16          // batch (graphs)
#define NN_   8192        // nodes per graph
#define EE_   131072      // edges per graph
#define HH_   128         // hidden / input channels
#define HL_   128         // head hidden
#define CC_   10          // classes
#define K1_   4096        // pool1 keep
#define K2_   2048        // pool2 keep

typedef __attribute__((ext_vector_type(2)))  float    v2f;
typedef __attribute__((ext_vector_type(8)))  float    v8f;
typedef __attribute__((ext_vector_type(16))) __bf16   v16bf;
typedef __attribute__((ext_vector_type(4)))  unsigned v4u;
typedef __attribute__((ext_vector_type(8)))  int      v8i;
typedef __attribute__((ext_vector_type(4)))  int      v4i;

// ---------------------------------------------------------------------------
// Generic fill
// ---------------------------------------------------------------------------
__global__ void fill_f32(float* __restrict__ p, float v, long long n) {
  long long id = blockIdx.x * (long long)blockDim.x + threadIdx.x;
  if (id < n) p[id] = v;
}

// ---------------------------------------------------------------------------
// GEMM: C[b] = A[b](rows x 128) @ W(128 x 128), fp32 WMMA.
//  * Workgroup = 256 threads (8 waves); block tile = 128 rows x 128 cols.
//  * W staged in LDS once per block via the Tensor Data Mover
//    (tensor_load_to_lds, D# built per ISA 8.3/8.4), s_wait_tensorcnt + barrier.
//  * Each wave owns 16 rows; 8 x (16x16) f32 accumulators; per K-step one
//    float2 A load feeds 8 v_wmma_f32_16x16x4_f32 with B fragments from LDS.
// Lane layouts per CDNA5 ISA 7.12.2:
//   A 16x4 : lanes 0-15 rows M=0..15; VGPR0 = K{0|2}, VGPR1 = K{1|3} (half sel)
//   B 4x16 : lanes 0-15 cols N=0..15; VGPR0 = K{0|2}, VGPR1 = K{1|3}
//   C 16x16: VGPR r -> (M = r + 8*half, N = lane&15)
// ---------------------------------------------------------------------------
__global__ void gemm_wmma(const float* __restrict__ A, const float* __restrict__ W,
                          float* __restrict__ Cout, int rows) {
  const int K = HH_, cols = HH_;
  __shared__ float sW[HH_ * HH_];  // 64 KB (WGP has 320 KB LDS)

#if __has_builtin(__builtin_amdgcn_tensor_load_to_lds)
  if (threadIdx.x < 32) {  // wave 0 issues the TDM DMA (EXEC ignored by TDM)
    const unsigned long long gaddr = (unsigned long long)(const void*)W;
    const unsigned ldsoff = (unsigned)(unsigned long long)(const void*)sW;
    v4u g0; v8i g1;
    v4i g2 = {0, 0, 0, 0};
    v4i g3 = {0, 0, 0, 0};
    v8i g4 = {0, 0, 0, 0, 0, 0, 0, 0};
    // Group 0: count=1 | lds_addr | global_addr(57b) | type=2
    g0[0] = 1u;                                      // count = 1 valid descriptor
    g0[1] = ldsoff;                                  // lds_addr [63:32]
    g0[2] = (unsigned)(gaddr & 0xFFFFFFFFu);         // global_addr lo
    g0[3] = ((unsigned)(gaddr >> 32) & 0x01FFFFFFu)  // global_addr hi (57-bit)
            | (2u << 30);                            // type = 2 ("image")
    // Group 1: data_size=4B; tensor 128x128; tile 128x128; dim0 stride 128
    g1[0] = (int)(2u << 16);                         // data_size = 2 -> 4 bytes
    g1[1] = (int)((unsigned)HH_ << 16);              // tensor_dim0[15:0] @ [31:16]
    g1[2] = (int)((unsigned)HH_ << 16);              // dim0 hi=0 | tensor_dim1[15:0]
    g1[3] = (int)((unsigned)HH_ << 16);              // dim1 hi=0 | tile_dim0 = 128
    g1[4] = (int)(unsigned)HH_;                      // tile_dim1 = 128 | tile_dim2=0
    g1[5] = (int)(unsigned)HH_;                      // tensor_dim0_stride lo = 128
    g1[6] = 0;                                       // stride hi | dim1_stride lo
    g1[7] = 0;
    // clang-23 / therock-10.0 form: 6 args (extra int32x8 group, then cpol)
    __builtin_amdgcn_tensor_load_to_lds(g0, g1, g2, g3, g4, 0);
    __builtin_amdgcn_s_wait_tensorcnt(0);
  }
#else
  for (int i = threadIdx.x; i < (HH_ * HH_) / 4; i += blockDim.x)
    ((float4*)sW)[i] = ((const float4*)W)[i];
#endif
  __syncthreads();

  const int lane = threadIdx.x & 31;   // wave32
  const int wave = threadIdx.x >> 5;   // 0..7
  const int half = lane >> 4;
  const int l16  = lane & 15;
  const int m0   = blockIdx.x * 128 + wave * 16;
  const float* Ag = A + (size_t)blockIdx.y * rows * K;
  float*       Cg = Cout + (size_t)blockIdx.y * rows * cols;

  v8f acc[8];
#pragma unroll
  for (int nt = 0; nt < 8; ++nt) acc[nt] = (v8f){};

  for (int k0 = 0; k0 < K; k0 += 4) {
    const int ka = k0 + half * 2;
    v2f a;
    a.x = Ag[(size_t)(m0 + l16) * K + ka];
    a.y = Ag[(size_t)(m0 + l16) * K + ka + 1];
#pragma unroll
    for (int nt = 0; nt < 8; ++nt) {
      const int n = nt * 16 + l16;
      v2f b;
      b.x = sW[(ka    ) * cols + n];
      b.y = sW[(ka + 1) * cols + n];
      acc[nt] = __builtin_amdgcn_wmma_f32_16x16x4_f32(
          false, a, false, b, (short)0, acc[nt], false, false);
    }
  }
#pragma unroll
  for (int nt = 0; nt < 8; ++nt) {
#pragma unroll
    for (int r = 0; r < 8; ++r) {
      const int row = m0 + r + half * 8;
      Cg[(size_t)row * cols + nt * 16 + l16] = acc[nt][r];
    }
  }
}

// ---------------------------------------------------------------------------
// Degree: deg = mask (self loop), then += mask[src]*mask[dst] per edge.
// ---------------------------------------------------------------------------
__global__ void deg_init(const float* __restrict__ mask, float* __restrict__ deg,
                         long long n) {
  long long id = blockIdx.x * (long long)blockDim.x + threadIdx.x;
  if (id < n) deg[id] = mask[id];
}

__global__ void deg_accum(const int* __restrict__ ei, const float* __restrict__ mask,
                          float* __restrict__ deg) {
  long long id = blockIdx.x * (long long)blockDim.x + threadIdx.x;
  if (id >= (long long)BB_ * EE_) return;
  const int b = (int)(id / EE_);
  const int e = (int)(id % EE_);
  const int* eib = ei + (size_t)b * 2 * EE_;
  const int s = eib[e], d = eib[EE_ + e];
  const float* mk = mask + (size_t)b * NN_;
  const float ew = mk[s] * mk[d];
  if (ew != 0.f) atomicAdd(&deg[(size_t)b * NN_ + d], ew);
}

__global__ void dinv_k(const float* __restrict__ deg, float* __restrict__ dinv,
                       long long n) {
  long long id = blockIdx.x * (long long)blockDim.x + threadIdx.x;
  if (id < n) {
    const float dg = deg[id];
    dinv[id] = (dg > 0.f) ? rsqrtf(dg) : 0.f;
  }
}

// ---------------------------------------------------------------------------
// Edge scatter: agg[dst] += feat[src] * norm.
//   useNorm=1 -> norm = dinv[s]*ew*dinv[d]  (GCN aggregation)
//   useNorm=0 -> norm = ew                  (SAGPool GraphConv aggregation)
// One thread per (b, edge, 4-channel chunk); float4 gather + 4 L2 atomics.
// ---------------------------------------------------------------------------
__global__ void scatter_edges(const int* __restrict__ ei, const float* __restrict__ feat,
                              const float* __restrict__ mask, const float* __restrict__ dinv,
                              float* __restrict__ agg, int useNorm) {
  const int CH = HH_ / 4;  // 32 chunks
  long long id = blockIdx.x * (long long)blockDim.x + threadIdx.x;
  if (id >= (long long)BB_ * EE_ * CH) return;
  const int  c4 = (int)(id % CH);
  long long  t  = id / CH;
  const int  e  = (int)(t % EE_);
  const int  b  = (int)(t / EE_);
  const int* eib = ei + (size_t)b * 2 * EE_;
  const int s = eib[e], d = eib[EE_ + e];
  const float* mk = mask + (size_t)b * NN_;
  float nrm = mk[s] * mk[d];
  if (nrm == 0.f) return;
  if (useNorm) {
    const float* dv = dinv + (size_t)b * NN_;
    nrm *= dv[s] * dv[d];
  }
  const float* fp = feat + ((size_t)b * NN_ + s) * HH_ + c4 * 4;
  __builtin_prefetch(fp, 0, 1);  // gfx1250 global_prefetch_b8
  const float4 v = *(const float4*)fp;
  float* op = agg + ((size_t)b * NN_ + d) * HH_ + c4 * 4;
  atomicAdd(op + 0, v.x * nrm);
  atomicAdd(op + 1, v.y * nrm);
  atomicAdd(op + 2, v.z * nrm);
  atomicAdd(op + 3, v.w * nrm);
}

// ---------------------------------------------------------------------------
// GCN finish: out = relu(agg + xw*dinv^2*mask + bias) * mask
// ---------------------------------------------------------------------------
__global__ void gcn_finish(const float* __restrict__ agg, const float* __restrict__ xw,
                           const float* __restrict__ dinv, const float* __restrict__ mask,
                           const float* __restrict__ bias, float* __restrict__ out) {
  long long id = blockIdx.x * (long long)blockDim.x + threadIdx.x;
  if (id >= (long long)BB_ * NN_ * HH_) return;
  const int       c  = (int)(id % HH_);
  const long long ni = id / HH_;  // b*N + i
  const float m  = mask[ni];
  const float dv = dinv[ni];
  const float v  = agg[id] + xw[id] * dv * dv * m + bias[c];
  out[id] = fmaxf(v, 0.f) * m;
}

// ---------------------------------------------------------------------------
// SAGPool score: score[i] = (agg2[i].wrel + h[i].wroot + b) * mask[i]
// One wave per node; lane handles 4 channels; wave32 xor-reduce.
// Also emits monotone uint key (0 for dead nodes) for radix select.
// ---------------------------------------------------------------------------
__global__ void score_k(const float* __restrict__ agg, const float* __restrict__ h,
                        const float* __restrict__ wrel, const float* __restrict__ wroot,
                        const float* __restrict__ bb, const float* __restrict__ mask,
                        float* __restrict__ score, unsigned* __restrict__ keys) {
  long long gid = blockIdx.x * (long long)blockDim.x + threadIdx.x;
  const int lane = (int)(gid & 31);
  long long w = gid >> 5;
  if (w >= (long long)BB_ * NN_) return;
  const int b = (int)(w / NN_);
  const int i = (int)(w % NN_);
  const size_t base = ((size_t)b * NN_ + i) * HH_ + lane * 4;
  float s = 0.f;
#pragma unroll
  for (int t = 0; t < 4; ++t) {
    const int c = lane * 4 + t;
    s += agg[base + t] * wrel[c] + h[base + t] * wroot[c];
  }
  for (int off = 16; off; off >>= 1) s += __shfl_xor(s, off, 32);
  if (lane == 0) {
    const size_t ni = (size_t)b * NN_ + i;
    const float  m  = mask[ni];
    const float  sc = (s + bb[0]) * m;
    score[ni] = sc;
    unsigned u = 0u;
    if (m > 0.f) {
      const unsigned bits = __float_as_uint(sc);
      u = (bits & 0x80000000u) ? ~bits : (bits | 0x80000000u);
    }
    keys[ni] = u;
  }
}

// ---------------------------------------------------------------------------
// Exact deterministic top-k via 4-pass MSB radix select on uint keys.
// ---------------------------------------------------------------------------
__global__ void sel_init(unsigned* __restrict__ prefix, int* __restrict__ krem, int k) {
  const int b = threadIdx.x;
  if (b < BB_) { prefix[b] = 0u; krem[b] = k; }
}

__global__ void hist_k(const unsigned* __restrict__ keys, const unsigned* __restrict__ prefix,
                       int* __restrict__ hist, int p) {
  __shared__ int sh[256];
  const int b = blockIdx.x, t = threadIdx.x;
  sh[t] = 0;
  __syncthreads();
  const unsigned pre = prefix[b];
  const unsigned hi  = (p == 3) ? 0u : (0xFFFFFFFFu << (unsigned)((p + 1) * 8));
  for (int i = t; i < NN_; i += 256) {
    const unsigned key = keys[(size_t)b * NN_ + i];
    if ((key & hi) == (pre & hi)) atomicAdd(&sh[(key >> (p * 8)) & 255], 1);
  }
  __syncthreads();
  hist[b * 256 + t] = sh[t];
}

__global__ void scan_k(const int* __restrict__ hist, unsigned* __restrict__ prefix,
                       int* __restrict__ krem, int p) {
  const int b = blockIdx.x;
  const int k = krem[b];
  const int* hb = hist + b * 256;
  long long cum = 0;
  for (int bin = 255; bin >= 0; --bin) {
    const int c = hb[bin];
    if (cum + c >= k) {
      prefix[b] |= ((unsigned)bin) << (p * 8);
      krem[b] = k - (int)cum;
      return;
    }
    cum += c;
  }
}

__global__ void make_mask(const unsigned* __restrict__ keys, const unsigned* __restrict__ prefix,
                          float* __restrict__ mask) {
  long long id = blockIdx.x * (long long)blockDim.x + threadIdx.x;
  if (id >= (long long)BB_ * NN_) return;
  const int b = (int)(id / NN_);
  mask[id] = (keys[id] > prefix[b]) ? 1.f : 0.f;
}

__global__ void tie_k(const unsigned* __restrict__ keys, const unsigned* __restrict__ prefix,
                      const int* __restrict__ krem, float* __restrict__ mask) {
  const int b = blockIdx.x;
  const unsigned thr = prefix[b];
  int rem = krem[b];
  for (int i = 0; i < NN_ && rem > 0; ++i) {
    const size_t ni = (size_t)b * NN_ + i;
    if (keys[ni] == thr) { mask[ni] = 1.f; --rem; }
  }
}

__global__ void gate_k(float* __restrict__ h, const float* __restrict__ score,
                       const float* __restrict__ mask) {
  long long id = blockIdx.x * (long long)blockDim.x + threadIdx.x;
  if (id >= (long long)BB_ * NN_ * HH_) return;
  const long long ni = id / HH_;
  h[id] *= tanhf(score[ni]) * mask[ni];
}

// ---------------------------------------------------------------------------
// Head: mean pool, lin1+relu, lin2, log_softmax
// ---------------------------------------------------------------------------
__global__ void meanpool_k(const float* __restrict__ h, float* __restrict__ g) {
  const int b = blockIdx.x, c = threadIdx.x;
  const float* p = h + (size_t)b * NN_ * HH_ + c;
  float s = 0.f;
  for (int i = 0; i < NN_; ++i) s += p[(size_t)i * HH_];
  g[b * HH_ + c] = s / (float)K2_;
}

__global__ void head1_k(const float* __restrict__ g, const float* __restrict__ w,
                        const float* __restrict__ bias, float* __restrict__ hh) {
  const int b = blockIdx.x, j = threadIdx.x;
  float s = bias[j];
  for (int c = 0; c < HH_; ++c) s += g[b * HH_ + c] * w[c * HL_ + j];
  hh[b * HL_ + j] = fmaxf(s, 0.f);
}

__global__ void head2_k(const float* __restrict__ hh, const float* __restrict__ w,
                        const float* __restrict__ bias, float* __restrict__ out) {
  __shared__ float lg[CC_];
  __shared__ float lse;
  const int b = blockIdx.x, j = threadIdx.x;
  if (j < CC_) {
    float s = bias[j];
    for (int c = 0; c < HL_; ++c) s += hh[b * HL_ + c] * w[c * CC_ + j];
    lg[j] = s;
  }
  __syncthreads();
  if (j == 0) {
    float m = lg[0];
    for (int t = 1; t < CC_; ++t) m = fmaxf(m, lg[t]);
    float s = 0.f;
    for (int t = 0; t < CC_; ++t) s += expf(lg[t] - m);
    lse = m + logf(s);
  }
  __syncthreads();
  if (j < CC_) out[b * CC_ + j] = lg[j] - lse;
}

// ---------------------------------------------------------------------------
// Host driver
// ---------------------------------------------------------------------------
extern "C" void kernel_launch(void* const* d_in, const int* in_sizes, int n_in,
                              void* d_out, int out_size, void* d_ws, size_t ws_size,
                              hipStream_t stream) {
  (void)in_sizes; (void)n_in; (void)out_size; (void)ws_size;

  const float* x    = (const float*)d_in[0];
  const int*   ei   = (const int*)d_in[1];
  const float* W1   = (const float*)d_in[2];
  const float* b1   = (const float*)d_in[3];
  const float* p1r  = (const float*)d_in[4];
  const float* p1o  = (const float*)d_in[5];
  const float* p1b  = (const float*)d_in[6];
  const float* W2   = (const float*)d_in[7];
  const float* b2   = (const float*)d_in[8];
  const float* p2r  = (const float*)d_in[9];
  const float* p2o  = (const float*)d_in[10];
  const float* p2b  = (const float*)d_in[11];
  const float* W3   = (const float*)d_in[12];
  const float* b3   = (const float*)d_in[13];
  const float* l1w  = (const float*)d_in[14];
  const float* l1b  = (const float*)d_in[15];
  const float* l2w  = (const float*)d_in[16];
  const float* l2b  = (const float*)d_in[17];
  float* out = (float*)d_out;

  // workspace carving (float-sized slots; ints/uints alias 4-byte slots)
  const long long FH = (long long)BB_ * NN_ * HH_;  // per-feature-buffer floats
  const long long BN = (long long)BB_ * NN_;
  float* ws = (float*)d_ws;
  float*    hA     = ws;
  float*    hB     = hA + FH;
  float*    xw     = hB + FH;
  float*    agg    = xw + FH;
  float*    deg    = agg + FH;
  float*    dinv   = deg + BN;
  float*    mask   = dinv + BN;
  float*    score  = mask + BN;
  unsigned* keys   = (unsigned*)(score + BN);
  int*      hist   = (int*)(keys + BN);
  unsigned* prefix = (unsigned*)(hist + BB_ * 256);
  int*      krem   = (int*)(prefix + BB_);
  float*    gbuf   = (float*)(krem + BB_);
  float*    hhbuf  = gbuf + (long long)BB_ * HH_;

  const int TPB = 256;
  const long long BNH     = FH;
  const int blkBN         = (int)((BN + TPB - 1) / TPB);
  const int blkBNH        = (int)((BNH + TPB - 1) / TPB);
  const int blkEdges      = (int)(((long long)BB_ * EE_ + TPB - 1) / TPB);
  const int blkScatter    = (int)(((long long)BB_ * EE_ * (HH_ / 4) + TPB - 1) / TPB);
  const int blkScore      = (int)(((long long)BB_ * NN_ * 32 + TPB - 1) / TPB);
  const dim3 gemmGrid(NN_ / 128, BB_);  // 128-row block tiles x batch

  auto gcn = [&](const float* hin, float* hout, const float* W, const float* bias) {
    deg_init<<<blkBN, TPB, 0, stream>>>(mask, deg, BN);
    deg_accum<<<blkEdges, TPB, 0, stream>>>(ei, mask, deg);
    dinv_k<<<blkBN, TPB, 0, stream>>>(deg, dinv, BN);
    gemm_wmma<<<gemmGrid, 256, 0, stream>>>(hin, W, xw, NN_);
    fill_f32<<<blkBNH, TPB, 0, stream>>>(agg, 0.f, BNH);
    scatter_edges<<<blkScatter, TPB, 0, stream>>>(ei, xw, mask, dinv, agg, 1);
    gcn_finish<<<blkBNH, TPB, 0, stream>>>(agg, xw, dinv, mask, bias, hout);
  };

  auto pool = [&](float* h, const float* wrel, const float* wroot, const float* pb, int k) {
    fill_f32<<<blkBNH, TPB, 0, stream>>>(agg, 0.f, BNH);
    scatter_edges<<<blkScatter, TPB, 0, stream>>>(ei, h, mask, dinv, agg, 0);
    score_k<<<blkScore, TPB, 0, stream>>>(agg, h, wrel, wroot, pb, mask, score, keys);
    sel_init<<<1, BB_, 0, stream>>>(prefix, krem, k);
    for (int p = 3; p >= 0; --p) {
      hist_k<<<BB_, 256, 0, stream>>>(keys, prefix, hist, p);
      scan_k<<<BB_, 1, 0, stream>>>(hist, prefix, krem, p);
    }
    make_mask<<<blkBN, TPB, 0, stream>>>(keys, prefix, mask);
    tie_k<<<BB_, 1, 0, stream>>>(keys, prefix, krem, mask);
    gate_k<<<blkBNH, TPB, 0, stream>>>(h, score, mask);
  };

  // pipeline
  fill_f32<<<blkBN, TPB, 0, stream>>>(mask, 1.f, BN);
  gcn(x, hA, W1, b1);
  pool(hA, p1r, p1o, p1b, K1_);
  gcn(hA, hB, W2, b2);
  pool(hB, p2r, p2o, p2b, K2_);
  gcn(hB, hA, W3, b3);
  meanpool_k<<<BB_, HH_, 0, stream>>>(hA, gbuf);
  head1_k<<<BB_, HL_, 0, stream>>>(gbuf, l1w, l1b, hhbuf);
  head2_k<<<BB_, 32, 0, stream>>>(hhbuf, l2w, l2b, out);
}